// PerceptualAttentionBlock_63273458204840
// MI455X (gfx1250) — compile-verified
//
#include <hip/hip_runtime.h>
#include <math.h>

// ---------------------------------------------------------------------------
// PerceptualAttentionBlock for MI455X (gfx1250, wave32).
// Key simplification: the hypernet "pool" is nearest-replication, so each
// QKV tensor has only (B,1024,2) distinct values -> Q/K/V rows are rank-2
// combos of precomputed weight half-sums, and attention runs on a 1024x1024
// half grid (rows repeat in pairs). GEMMs use V_WMMA_F32_16X16X4_F32 to keep
// full fp32 precision (total FLOPs ~4.3G: latency-bound, precision-critical).
// ---------------------------------------------------------------------------

typedef __attribute__((ext_vector_type(2))) float v2f;
typedef __attribute__((ext_vector_type(8))) float v8f;

#define NB 2        // batches
#define LFULL 2048
#define NH 1024     // half grid
#define DD 512
#define HID 32

// workspace layout (float offsets)
#define OFF_H     ((size_t)0)                       // 2*1024*2      = 4096
#define OFF_G     ((size_t)4096)                    // 3*2*1024*2    = 12288
#define OFF_WSUM  ((size_t)16384)                   // 3*1024        = 3072
#define OFF_Q     ((size_t)20480)                   // 2*1024*512
#define OFF_K     (OFF_Q + (size_t)NB*NH*DD)
#define OFF_V     (OFF_K + (size_t)NB*NH*DD)
#define OFF_SC    (OFF_V + (size_t)NB*NH*DD)        // 2*1024*1024
#define OFF_SROW  (OFF_SC + (size_t)NB*NH*NH)       // 2048

// ---------------- k_hash: h[b][m][c] = 0.5*(proj[2m]+proj[2m+1]) -----------
__global__ __launch_bounds__(32) void k_hash(const float* __restrict__ x,
                                             const float* __restrict__ hw,
                                             const float* __restrict__ hb,
                                             float* __restrict__ h) {
  const int m = blockIdx.x, b = blockIdx.y, lane = threadIdx.x;
  const float* r0 = x + ((size_t)b * LFULL + 2 * m) * DD;
  const float* r1 = r0 + DD;
  float a00 = 0.f, a01 = 0.f, a10 = 0.f, a11 = 0.f;
  for (int d = lane; d < DD; d += 32) {
    float w0 = hw[d * 2 + 0], w1 = hw[d * 2 + 1];
    float x0 = r0[d], x1 = r1[d];
    a00 += x0 * w0; a01 += x0 * w1;
    a10 += x1 * w0; a11 += x1 * w1;
  }
  for (int off = 16; off; off >>= 1) {
    a00 += __shfl_down(a00, off, 32);
    a01 += __shfl_down(a01, off, 32);
    a10 += __shfl_down(a10, off, 32);
    a11 += __shfl_down(a11, off, 32);
  }
  if (lane == 0) {
    h[((size_t)b * NH + m) * 2 + 0] = 0.5f * (a00 + a10) + hb[0];
    h[((size_t)b * NH + m) * 2 + 1] = 0.5f * (a01 + a11) + hb[1];
  }
}

// ---------------- k_wsum: per-row half sums of qw/kw/vw --------------------
__global__ __launch_bounds__(32) void k_wsum(const float* __restrict__ qw,
                                             const float* __restrict__ kw,
                                             const float* __restrict__ vw,
                                             float* __restrict__ wsum) {
  const int o = blockIdx.x, j = blockIdx.y, lane = threadIdx.x;
  const float* w = ((j == 0) ? qw : (j == 1) ? kw : vw) + (size_t)o * DD;
  float s0 = 0.f, s1 = 0.f;
  for (int d = lane; d < 256; d += 32) { s0 += w[d]; s1 += w[d + 256]; }
  for (int off = 16; off; off >>= 1) {
    s0 += __shfl_down(s0, off, 32);
    s1 += __shfl_down(s1, off, 32);
  }
  if (lane == 0) {
    wsum[(size_t)j * 1024 + o]       = s0;
    wsum[(size_t)j * 1024 + 512 + o] = s1;
  }
}

// ---------------- k_hyper: conv3x3 -> conv3x3 -> 1x1 -> 1x1 in LDS ---------
__global__ __launch_bounds__(256) void k_hyper(const float* __restrict__ h,
    const float* __restrict__ e1w, const float* __restrict__ e1b,
    const float* __restrict__ e2w, const float* __restrict__ e2b,
    const float* __restrict__ d1w, const float* __restrict__ d1b,
    const float* __restrict__ d2w, const float* __restrict__ d2b,
    float* __restrict__ g) {
  const int tid = threadIdx.x;
  const int tm = blockIdx.x * 64;   // 16 tiles of 64 rows
  const int i  = blockIdx.y;        // which hypernet (Q/K/V)
  const int b  = blockIdx.z;

  __shared__ float sh_h[68 * 2];        // h tile, halo 2 each side
  __shared__ float sw1[288];            // e1w[i]
  __shared__ float sb1[32];
  __shared__ float sw2[9216];           // e2w[i]
  __shared__ float sb2[32];
  __shared__ float sd1w[1024];
  __shared__ float sd1b[32];
  __shared__ float sd2w[32];
  __shared__ float sd2b_s[1];
  __shared__ float sc1[32 * 66 * 2];    // c1 with 1-row halo
  __shared__ float sc2[32 * 64 * 2];

  for (int idx = tid; idx < 136; idx += 256) {
    int r = idx >> 1, w = idx & 1;
    int m = tm - 2 + r;
    sh_h[idx] = (m >= 0 && m < NH) ? h[((size_t)b * NH + m) * 2 + w] : 0.f;
  }
  for (int idx = tid; idx < 288; idx += 256)  sw1[idx]  = e1w[(size_t)i * 288 + idx];
  for (int idx = tid; idx < 9216; idx += 256) sw2[idx]  = e2w[(size_t)i * 9216 + idx];
  for (int idx = tid; idx < 1024; idx += 256) sd1w[idx] = d1w[(size_t)i * 1024 + idx];
  if (tid < 32) {
    sb1[tid]  = e1b[i * 32 + tid];
    sb2[tid]  = e2b[i * 32 + tid];
    sd1b[tid] = d1b[i * 32 + tid];
    sd2w[tid] = d2w[i * 32 + tid];
  }
  if (tid == 0) sd2b_s[0] = d2b[i];
  __syncthreads();

  // conv1: 1 -> 32 channels, outputs rows [tm-1, tm+64] (66 rows, 1-row halo)
  for (int idx = tid; idx < 32 * 66 * 2; idx += 256) {
    int ch = idx / 132, rem = idx % 132;
    int r = rem >> 1, w = rem & 1;
    int m = tm - 1 + r;
    if (m >= 0 && m < NH) {
      float val = sb1[ch];
      #pragma unroll
      for (int dy = 0; dy < 3; ++dy)
        #pragma unroll
        for (int dx = 0; dx < 3; ++dx) {
          int ww = w + dx - 1;
          if (ww >= 0 && ww < 2)
            val += sw1[ch * 9 + dy * 3 + dx] * sh_h[(r + dy) * 2 + ww];
        }
      sc1[idx] = fmaxf(val, 0.f);
    } else {
      sc1[idx] = 0.f;  // SAME-padding for conv2
    }
  }
  __syncthreads();

  // conv2: 32 -> 32 channels, rows [tm, tm+63]
  for (int idx = tid; idx < 32 * 64 * 2; idx += 256) {
    int co = idx >> 7, rem = idx & 127;
    int r2 = rem >> 1, w = rem & 1;
    float val = sb2[co];
    for (int ci = 0; ci < 32; ++ci) {
      const float* wp = &sw2[(co * 32 + ci) * 9];
      const float* cp = &sc1[ci * 132];
      #pragma unroll
      for (int dy = 0; dy < 3; ++dy) {
        int base = (r2 + dy) * 2;
        if (w == 0) {
          val += wp[dy * 3 + 1] * cp[base + 0];
          val += wp[dy * 3 + 2] * cp[base + 1];
        } else {
          val += wp[dy * 3 + 0] * cp[base + 0];
          val += wp[dy * 3 + 1] * cp[base + 1];
        }
      }
    }
    sc2[idx] = fmaxf(val, 0.f);
  }
  __syncthreads();

  // dec1 (1x1, relu) + dec2 (1x1): one scalar per (m, w)
  if (tid < 128) {
    int r2 = tid >> 1, w = tid & 1;
    float gacc = sd2b_s[0];
    for (int co = 0; co < 32; ++co) {
      float t = sd1b[co];
      for (int ci = 0; ci < 32; ++ci)
        t += sd1w[co * 32 + ci] * sc2[ci * 128 + r2 * 2 + w];
      gacc += sd2w[co] * fmaxf(t, 0.f);
    }
    g[(((size_t)i * NB + b) * NH + tm + r2) * 2 + w] = gacc;
  }
}

// ---------------- k_qkv: rank-2 projection + l2norm, half-grid rows --------
__global__ __launch_bounds__(128) void k_qkv(const float* __restrict__ g,
    const float* __restrict__ wsum,
    const float* __restrict__ qb, const float* __restrict__ kb,
    const float* __restrict__ vb,
    float* __restrict__ Qh, float* __restrict__ Kh, float* __restrict__ Vh) {
  const int m = blockIdx.x, b = blockIdx.y, tid = threadIdx.x;
  __shared__ float red[4];
  #pragma unroll
  for (int j = 0; j < 3; ++j) {
    const float* bias = (j == 0) ? qb : (j == 1) ? kb : vb;
    float* outp       = (j == 0) ? Qh : (j == 1) ? Kh : Vh;
    const float u = g[(((size_t)j * NB + b) * NH + m) * 2 + 0];
    const float v = g[(((size_t)j * NB + b) * NH + m) * 2 + 1];
    float t[4];
    float ss = 0.f;
    #pragma unroll
    for (int e = 0; e < 4; ++e) {
      int o = tid + e * 128;
      float val = u * wsum[(size_t)j * 1024 + o] +
                  v * wsum[(size_t)j * 1024 + 512 + o] + bias[o];
      t[e] = val;
      ss += val * val;
    }
    for (int off = 16; off; off >>= 1) ss += __shfl_down(ss, off, 32);
    if ((tid & 31) == 0) red[tid >> 5] = ss;
    __syncthreads();
    float tot = red[0] + red[1] + red[2] + red[3];
    float inv = 1.f / fmaxf(sqrtf(tot), 1e-12f);
    #pragma unroll
    for (int e = 0; e < 4; ++e)
      outp[((size_t)b * NH + m) * DD + tid + e * 128] = t[e] * inv;
    __syncthreads();
  }
}

// ---------------- k_gemm_qk: scores = f(Qh @ Kh^T / sqrt(D)) via WMMA f32 --
__global__ __launch_bounds__(32) void k_gemm_qk(const float* __restrict__ Qh,
                                                const float* __restrict__ Kh,
                                                float* __restrict__ sc) {
  const int tn = blockIdx.x << 4;
  const int tm = blockIdx.y << 4;
  const int b  = blockIdx.z;
  const int lane = threadIdx.x;
  const int row  = lane & 15;
  const int kh   = (lane >> 4) << 1;  // K offset {0,2} per lane half
  const float* qp = Qh + ((size_t)b * NH + tm + row) * DD + kh;
  const float* kp = Kh + ((size_t)b * NH + tn + row) * DD + kh;
  v8f acc = {0.f, 0.f, 0.f, 0.f, 0.f, 0.f, 0.f, 0.f};
  for (int k0 = 0; k0 < DD; k0 += 4) {
    v2f a  = *(const v2f*)(qp + k0);
    v2f bf = *(const v2f*)(kp + k0);
    acc = __builtin_amdgcn_wmma_f32_16x16x4_f32(false, a, false, bf,
                                                (short)0, acc, false, false);
  }
  const float inv_sqrt_d = 0.04419417382415922f;  // 1/sqrt(512)
  const int n  = row;
  const int mb = (lane >> 4) << 3;  // C: lanes>=16 hold M = r + 8
  float* op = sc + ((size_t)b * NH + tm + mb) * NH + tn + n;
  #pragma unroll
  for (int r = 0; r < 8; ++r) {
    float logit = acc[r] * inv_sqrt_d;
    float xx = 0.5f * logit - 0.2f;                      // (alpha-1)*logit - tau
    float sp = (xx > 0.f) ? (xx + log1pf(expf(-xx))) : log1pf(expf(xx));
    op[(size_t)r * NH] = sp * sp;                        // ^(1/(alpha-1)) = ^2
  }
}

// ---------------- k_rowsum: S[row] = sum_k scores[row][k] ------------------
__global__ __launch_bounds__(256) void k_rowsum(const float* __restrict__ sc,
                                                float* __restrict__ srow) {
  const int row  = blockIdx.x * 8 + (threadIdx.x >> 5);  // 2048 rows
  const int lane = threadIdx.x & 31;
  const float* p = sc + (size_t)row * NH;
  float s = 0.f;
  for (int k = lane; k < NH; k += 32) s += p[k];
  for (int off = 16; off; off >>= 1) s += __shfl_down(s, off, 32);
  if (lane == 0) srow[row] = s;
}

// ---------------- k_gemm_ov: O = scores @ Vh via WMMA f32, expand rows -----
__global__ __launch_bounds__(32) void k_gemm_ov(const float* __restrict__ sc,
                                                const float* __restrict__ Vh,
                                                const float* __restrict__ srow,
                                                float* __restrict__ out) {
  const int tn = blockIdx.x << 4;   // 32 tiles over D=512
  const int tm = blockIdx.y << 4;
  const int b  = blockIdx.z;
  const int lane = threadIdx.x;
  const int row  = lane & 15;
  const int kh   = (lane >> 4) << 1;
  const float* ap = sc + ((size_t)b * NH + tm + row) * NH + kh;
  const float* vp = Vh + (size_t)b * NH * DD + tn + row;
  v8f acc = {0.f, 0.f, 0.f, 0.f, 0.f, 0.f, 0.f, 0.f};
  for (int k0 = 0; k0 < NH; k0 += 4) {
    v2f a = *(const v2f*)(ap + k0);
    v2f bf;
    bf.x = vp[(size_t)(k0 + kh) * DD];
    bf.y = vp[(size_t)(k0 + kh + 1) * DD];
    acc = __builtin_amdgcn_wmma_f32_16x16x4_f32(false, a, false, bf,
                                                (short)0, acc, false, false);
  }
  const int n  = row;
  const int mb = (lane >> 4) << 3;
  #pragma unroll
  for (int r = 0; r < 8; ++r) {
    int m = tm + mb + r;
    float S = srow[(size_t)b * NH + m];
    float f = 2.f / (2.f * S + 1e-6f);   // weights = score/(S_full+eps); V rows doubled
    float o = acc[r] * f;
    float* q0 = out + ((size_t)b * LFULL + 2 * m) * DD + tn + n;
    q0[0]  = o;
    q0[DD] = o;                          // Q rows repeat in pairs
  }
}

// ---------------- k_score: min-max normalized attention score --------------
__global__ __launch_bounds__(1024) void k_score(const float* __restrict__ srow,
                                                float* __restrict__ outS) {
  const int b = blockIdx.x, tid = threadIdx.x;
  __shared__ float smin[32], smax[32];
  const float S = srow[(size_t)b * NH + tid];
  const float raw = (2.f * S) / (2.f * S + 1e-6f);
  float mn = raw, mx = raw;
  for (int off = 16; off; off >>= 1) {
    mn = fminf(mn, __shfl_down(mn, off, 32));
    mx = fmaxf(mx, __shfl_down(mx, off, 32));
  }
  if ((tid & 31) == 0) { smin[tid >> 5] = mn; smax[tid >> 5] = mx; }
  __syncthreads();
  if (tid < 32) {
    mn = smin[tid]; mx = smax[tid];
    for (int off = 16; off; off >>= 1) {
      mn = fminf(mn, __shfl_down(mn, off, 32));
      mx = fmaxf(mx, __shfl_down(mx, off, 32));
    }
    if (tid == 0) { smin[0] = mn; smax[0] = mx; }
  }
  __syncthreads();
  mn = smin[0]; mx = smax[0];
  const float v = (raw - mn) / (mx - mn + 1e-6f);
  outS[(size_t)b * LFULL + 2 * tid]     = v;
  outS[(size_t)b * LFULL + 2 * tid + 1] = v;
}

// ---------------------------------------------------------------------------
extern "C" void kernel_launch(void* const* d_in, const int* in_sizes, int n_in,
                              void* d_out, int out_size, void* d_ws, size_t ws_size,
                              hipStream_t stream) {
  const float* x      = (const float*)d_in[0];
  const float* hash_w = (const float*)d_in[1];
  const float* hash_b = (const float*)d_in[2];
  const float* enc1_w = (const float*)d_in[3];
  const float* enc1_b = (const float*)d_in[4];
  const float* enc2_w = (const float*)d_in[5];
  const float* enc2_b = (const float*)d_in[6];
  const float* dec1_w = (const float*)d_in[7];
  const float* dec1_b = (const float*)d_in[8];
  const float* dec2_w = (const float*)d_in[9];
  const float* dec2_b = (const float*)d_in[10];
  const float* qw = (const float*)d_in[11];
  const float* qb = (const float*)d_in[12];
  const float* kw = (const float*)d_in[13];
  const float* kb = (const float*)d_in[14];
  const float* vw = (const float*)d_in[15];
  const float* vb = (const float*)d_in[16];

  float* ws  = (float*)d_ws;
  float* out = (float*)d_out;

  k_hash<<<dim3(NH, NB), 32, 0, stream>>>(x, hash_w, hash_b, ws + OFF_H);
  k_wsum<<<dim3(DD, 3), 32, 0, stream>>>(qw, kw, vw, ws + OFF_WSUM);
  k_hyper<<<dim3(16, 3, NB), 256, 0, stream>>>(ws + OFF_H, enc1_w, enc1_b,
      enc2_w, enc2_b, dec1_w, dec1_b, dec2_w, dec2_b, ws + OFF_G);
  k_qkv<<<dim3(NH, NB), 128, 0, stream>>>(ws + OFF_G, ws + OFF_WSUM,
      qb, kb, vb, ws + OFF_Q, ws + OFF_K, ws + OFF_V);
  k_gemm_qk<<<dim3(NH / 16, NH / 16, NB), 32, 0, stream>>>(
      ws + OFF_Q, ws + OFF_K, ws + OFF_SC);
  k_rowsum<<<dim3(NB * NH / 8), 256, 0, stream>>>(ws + OFF_SC, ws + OFF_SROW);
  k_gemm_ov<<<dim3(DD / 16, NH / 16, NB), 32, 0, stream>>>(
      ws + OFF_SC, ws + OFF_V, ws + OFF_SROW, out);
  k_score<<<dim3(NB), 1024, 0, stream>>>(ws + OFF_SROW,
      out + (size_t)NB * LFULL * DD);
}